// DiscriminativeLoss_22711787061841
// MI455X (gfx1250) — compile-verified
//
#include <hip/hip_runtime.h>
#include <hip/hip_bf16.h>
#include <math.h>

// Problem constants (fixed by setup_inputs)
constexpr int BS  = 8;
constexpr int F   = 16;     // n_filters
constexpr int NI  = 24;     // n_inst
constexpr int NIP = 32;     // padded instances (2 WMMA N-tiles)
constexpr int L   = 128 * 128;

constexpr float DELTA_V = 0.5f;
constexpr float MARGIN  = 2.0f * 1.5f;   // 2*DELTA_D
constexpr float GAMMA   = 0.001f;

// Workspace layout (float offsets)
constexpr size_t OFF_COUNTS = 0;                         // BS*NIP
constexpr size_t OFF_SUMS   = OFF_COUNTS + BS * NIP;     // BS*NIP*F
constexpr size_t OFF_MEANS  = OFF_SUMS + BS * NIP * F;   // BS*NIP*F (padded, zeros invalid)
constexpr size_t OFF_MNSQ   = OFF_MEANS + BS * NIP * F;  // BS*NIP
constexpr size_t OFF_VARNUM = OFF_MNSQ + BS * NIP;       // BS
constexpr size_t OFF_VARDEN = OFF_VARNUM + BS;           // BS
constexpr size_t OFF_DIST   = OFF_VARDEN + BS;           // BS
constexpr size_t OFF_REG    = OFF_DIST + BS;             // BS
constexpr size_t OFF_PNSQ   = OFF_REG + BS;              // BS*L

typedef __attribute__((ext_vector_type(2))) float v2f;
typedef __attribute__((ext_vector_type(8))) float v8f;

__device__ __forceinline__ float blockReduce256(float v, float* red) {
    int tid = threadIdx.x;
    red[tid] = v;
    __syncthreads();
    #pragma unroll
    for (int st = 128; st > 0; st >>= 1) {
        if (tid < st) red[tid] += red[tid + st];
        __syncthreads();
    }
    float r = red[0];
    __syncthreads();
    return r;
}

// Kernel 1: per (b,i) -> counts, per-filter masked sums (the 'bli,blf->bif' einsum)
__global__ void dl_stats(const float* __restrict__ input,
                         const int* __restrict__ target,
                         float* __restrict__ ws) {
    __shared__ float red[256];
    int i = blockIdx.x;   // 0..23
    int b = blockIdx.y;   // 0..7
    const int*   tg  = target + ((size_t)b * NI + i) * L;
    const float* inb = input + (size_t)b * F * L;

    float cnt = 0.0f;
    float s[F];
    #pragma unroll
    for (int f = 0; f < F; ++f) s[f] = 0.0f;

    for (int l = threadIdx.x; l < L; l += blockDim.x) {
        int t = tg[l];
        if (t) {
            float tf = (float)t;
            cnt += tf;
            #pragma unroll
            for (int f = 0; f < F; ++f) s[f] += tf * inb[f * L + l];
        }
    }

    float tot = blockReduce256(cnt, red);
    if (threadIdx.x == 0) ws[OFF_COUNTS + b * NIP + i] = tot;
    #pragma unroll
    for (int f = 0; f < F; ++f) {
        float sf = blockReduce256(s[f], red);
        if (threadIdx.x == 0) ws[OFF_SUMS + ((size_t)b * NIP + i) * F + f] = sf;
    }
}

// Kernel 2: per-pixel ||pred||^2
__global__ void dl_pnormsq(const float* __restrict__ input, float* __restrict__ ws) {
    int b = blockIdx.y;
    int l = blockIdx.x * blockDim.x + threadIdx.x;
    if (l >= L) return;
    const float* inb = input + (size_t)b * F * L;
    float ss = 0.0f;
    #pragma unroll
    for (int f = 0; f < F; ++f) {
        float v = inb[f * L + l];
        ss += v * v;
    }
    ws[OFF_PNSQ + (size_t)b * L + l] = ss;
}

// Kernel 3: per-batch means (padded), ||m||^2, var_den, dist term, reg term; zero var accum
__global__ void dl_means(const int* __restrict__ nobj_arr, float* __restrict__ ws) {
    __shared__ float red[256];
    int b   = blockIdx.x;
    int tid = threadIdx.x;
    int nobj = nobj_arr[b];
    float nf = (float)nobj;

    float* counts = ws + OFF_COUNTS + b * NIP;
    float* sums   = ws + OFF_SUMS   + (size_t)b * NIP * F;
    float* means  = ws + OFF_MEANS  + (size_t)b * NIP * F;
    float* mnsq   = ws + OFF_MNSQ   + b * NIP;

    // Padded means: zero for i >= nobj or i >= NI
    for (int idx = tid; idx < NIP * F; idx += blockDim.x) {
        int i = idx >> 4, f = idx & 15;
        float m = 0.0f;
        if (i < NI && i < nobj) {
            float c = counts[i];
            float denom = (c > 0.0f) ? c : 1.0f;
            m = sums[i * F + f] / denom;
        }
        means[i * F + f] = m;
    }
    __syncthreads();

    if (tid < NIP) {
        float ss = 0.0f;
        #pragma unroll
        for (int f = 0; f < F; ++f) {
            float m = means[tid * F + f];
            ss += m * m;
        }
        mnsq[tid] = ss;
    }
    __syncthreads();

    // var_den = sum of valid counts
    float vd = (tid < NI && tid < nobj) ? counts[tid] : 0.0f;
    vd = blockReduce256(vd, red);

    // pairwise hinge distance term
    float acc = 0.0f;
    for (int p = tid; p < NI * NI; p += blockDim.x) {
        int i = p / NI, j = p % NI;
        if (i != j && i < nobj && j < nobj) {
            float ss = 0.0f;
            #pragma unroll
            for (int f = 0; f < F; ++f) {
                float d = means[i * F + f] - means[j * F + f];
                ss += d * d;
            }
            float dn = (ss > 0.0f) ? sqrtf(ss) : 0.0f;
            float h = fmaxf(MARGIN - dn, 0.0f);
            acc += h * h;
        }
    }
    acc = blockReduce256(acc, red);

    // reg term: sum of valid row norms / nf
    float rv = (tid < NI && tid < nobj) ? sqrtf(mnsq[tid]) : 0.0f;
    rv = blockReduce256(rv, red);

    if (tid == 0) {
        ws[OFF_VARDEN + b] = vd;
        ws[OFF_DIST + b]   = (nobj > 1) ? (acc / (nf * (nf - 1.0f))) : 0.0f;
        ws[OFF_REG + b]    = rv / nf;
        ws[OFF_VARNUM + b] = 0.0f;
    }
}

// Kernel 4 (WMMA): cross = pred @ means^T via v_wmma_f32_16x16x4_f32, then
// hinge variance term: sum over (l,i) of max(||p_l - m_i|| - dv, 0)^2 * gt * valid.
__global__ void dl_var(const float* __restrict__ input,
                       const int* __restrict__ target,
                       const int* __restrict__ nobj_arr,
                       float* __restrict__ ws) {
    __shared__ float red[256];
    int b    = blockIdx.z;
    int i0   = blockIdx.y * 16;             // N tile: 0 or 16
    int wave = threadIdx.x >> 5;
    int lane = threadIdx.x & 31;
    int l0   = (blockIdx.x * 8 + wave) * 16; // M tile base (16 pixels)
    int r    = lane & 15;
    int kh   = lane >> 4;

    const float* inb = input + (size_t)b * F * L;
    const float* mb  = ws + OFF_MEANS + (size_t)b * NIP * F;

    // A: pred tile 16x16 (MxK) fed as 4 chunks of 16x4
    // B: means^T tile 16x16 (KxN) fed as 4 chunks of 4x16
    v8f c = {};
    #pragma unroll
    for (int kc = 0; kc < 4; ++kc) {
        int f0 = kc * 4 + 2 * kh;
        v2f a, bb;
        a.x  = inb[(size_t)f0 * L + l0 + r];
        a.y  = inb[(size_t)(f0 + 1) * L + l0 + r];
        bb.x = mb[(i0 + r) * F + f0];
        bb.y = mb[(i0 + r) * F + f0 + 1];
        c = __builtin_amdgcn_wmma_f32_16x16x4_f32(
                false, a, false, bb, (short)0, c, false, false);
    }
    // c[j] = cross[l0 + j + 8*kh, i0 + r]

    int i = i0 + r;
    int nobj = nobj_arr[b];
    bool valid = (i < NI) && (i < nobj);
    float mn = ws[OFF_MNSQ + b * NIP + i];
    const int* tg = valid ? (target + ((size_t)b * NI + i) * L) : nullptr;
    const float* pn = ws + OFF_PNSQ + (size_t)b * L;

    float acc = 0.0f;
    #pragma unroll
    for (int j = 0; j < 8; ++j) {
        int l = l0 + j + 8 * kh;
        float t = valid ? (float)tg[l] : 0.0f;
        if (t != 0.0f) {
            float ssq = fmaxf(pn[l] - 2.0f * c[j] + mn, 0.0f);
            float dn  = sqrtf(ssq);
            float h   = fmaxf(dn - DELTA_V, 0.0f);
            acc += t * h * h;
        }
    }
    acc = blockReduce256(acc, red);
    if (threadIdx.x == 0) atomicAdd(&ws[OFF_VARNUM + b], acc);
}

// Kernel 5: combine terms into scalar loss
__global__ void dl_final(const float* __restrict__ ws, float* __restrict__ out) {
    if (threadIdx.x == 0 && blockIdx.x == 0) {
        float loss = 0.0f;
        for (int b = 0; b < BS; ++b) {
            loss += ws[OFF_VARNUM + b] / ws[OFF_VARDEN + b];
            loss += ws[OFF_DIST + b];
            loss += GAMMA * ws[OFF_REG + b];
        }
        out[0] = loss / (float)BS;
    }
}

extern "C" void kernel_launch(void* const* d_in, const int* in_sizes, int n_in,
                              void* d_out, int out_size, void* d_ws, size_t ws_size,
                              hipStream_t stream) {
    const float* input  = (const float*)d_in[0];
    const int*   target = (const int*)d_in[1];
    const int*   nobj   = (const int*)d_in[2];
    float* ws  = (float*)d_ws;
    float* out = (float*)d_out;

    dl_stats<<<dim3(NI, BS), 256, 0, stream>>>(input, target, ws);
    dl_pnormsq<<<dim3(L / 256, BS), 256, 0, stream>>>(input, ws);
    dl_means<<<BS, 256, 0, stream>>>(nobj, ws);
    // 8 waves/block, each wave = one 16x16 WMMA output tile
    dl_var<<<dim3(L / 16 / 8, NIP / 16, BS), 256, 0, stream>>>(input, target, nobj, ws);
    dl_final<<<1, 32, 0, stream>>>(ws, out);
}